// GCN_11433202942732
// MI455X (gfx1250) — compile-verified
//
#include <hip/hip_runtime.h>

#define N_NODES 100000
#define N_EDGES 1600000
#define IN_DIM  64
#define HID_DIM 32
#define N_TILES (N_NODES / 16)   // 6250, exact
#define WT_STRIDE 68             // IN_DIM + 4 pad: conflict-free B reads

typedef __attribute__((ext_vector_type(2))) float v2f;
typedef __attribute__((ext_vector_type(8))) float v8f;

// ---------------------------------------------------------------------------
// Kernel 1: zero the output accumulator and both degree arrays
// ---------------------------------------------------------------------------
__global__ void gcn_zero_kernel(float* __restrict__ out, float* __restrict__ degs) {
    int i = blockIdx.x * blockDim.x + threadIdx.x;
    const int nOut = N_NODES * HID_DIM;
    if (i < nOut) {
        out[i] = 0.0f;
    } else if (i < nOut + 2 * N_NODES) {
        degs[i - nOut] = 0.0f;
    }
}

// ---------------------------------------------------------------------------
// Kernel 2: degree computation (atomic f32 adds; resolve in L2)
// ---------------------------------------------------------------------------
__global__ void gcn_degree_kernel(const int* __restrict__ src,
                                  const int* __restrict__ dst,
                                  float* __restrict__ deg_out,
                                  float* __restrict__ deg_in) {
    int e = blockIdx.x * blockDim.x + threadIdx.x;
    if (e >= N_EDGES) return;
    atomicAdd(&deg_out[src[e]], 1.0f);
    atomicAdd(&deg_in[dst[e]], 1.0f);
}

// ---------------------------------------------------------------------------
// Kernel 3: h = (X @ W) * rsqrt(max(deg_out,1))  via V_WMMA_F32_16X16X4_F32
// One wave per 16-row tile; two 16x16 N-tiles (HID_DIM=32); K=64 in 16 steps.
// A layout (16x4 f32): lane 0-15 -> M=lane, VGPRs {K0,K1}; lane 16-31 -> {K2,K3}.
// B layout (4x16 f32): VGPR0: lanes0-15=K0 row, lanes16-31=K2; VGPR1: K1/K3.
// C/D layout: VGPR v: lanes0-15 -> M=v, lanes16-31 -> M=v+8; N = lane&15.
// W is stored TRANSPOSED in LDS (WlT[col][k], stride 68) so each B fragment is
// one contiguous 8-byte ds_load_b64 into an even-aligned VGPR pair (no repack
// movs), and the 68-float stride makes the 32 lanes hit 32 distinct banks.
// ---------------------------------------------------------------------------
__global__ void __launch_bounds__(256)
gcn_gemm_wmma(const float* __restrict__ X,
              const float* __restrict__ W,
              const float* __restrict__ deg_out,
              float* __restrict__ h) {
    __shared__ float WlT[HID_DIM * WT_STRIDE];   // 8.5 KB of the 320KB LDS
    for (int i = threadIdx.x; i < IN_DIM * HID_DIM; i += 256) {
        const int k = i >> 5;        // row of W    (IN_DIM direction)
        const int c = i & 31;        // column of W (HID_DIM direction)
        WlT[c * WT_STRIDE + k] = W[i];
    }
    __syncthreads();

    const int lane = threadIdx.x & 31;
    const int wave = threadIdx.x >> 5;
    const int tile = blockIdx.x * 8 + wave;
    if (tile >= N_TILES) return;             // wave-uniform: EXEC stays all-1s
    const int rowBase = tile * 16;

    const int m     = lane & 15;             // row within tile (A) / column (B,C,D)
    const int khalf = (lane >> 4) << 1;      // 0 for lanes 0-15, 2 for lanes 16-31
    const float* xrow = X + (long)(rowBase + m) * IN_DIM + khalf;
    const float* wcol0 = WlT + m * WT_STRIDE + khalf;          // N-tile 0
    const float* wcol1 = WlT + (m + 16) * WT_STRIDE + khalf;   // N-tile 1

    v8f acc0 = {};
    v8f acc1 = {};
#pragma unroll
    for (int k = 0; k < IN_DIM; k += 4) {
        // A fragment: two consecutive f32 of this row (8B-aligned global b64)
        v2f a = *(const v2f*)(xrow + k);
        // B fragments: contiguous 8B ds_load_b64, conflict-free banking
        v2f b0 = *(const v2f*)(wcol0 + k);
        v2f b1 = *(const v2f*)(wcol1 + k);
        acc0 = __builtin_amdgcn_wmma_f32_16x16x4_f32(
            false, a, false, b0, (short)0, acc0, false, false);
        acc1 = __builtin_amdgcn_wmma_f32_16x16x4_f32(
            false, a, false, b1, (short)0, acc1, false, false);
    }

    // Fused norm_src scale + store, per the C/D VGPR layout
    const int mOff = (lane >> 4) << 3;       // 0 or 8
#pragma unroll
    for (int v = 0; v < 8; ++v) {
        const int r = rowBase + mOff + v;
        const float s = __frsqrt_rn(fmaxf(deg_out[r], 1.0f));
        h[(long)r * HID_DIM + m]      = acc0[v] * s;
        h[(long)r * HID_DIM + 16 + m] = acc1[v] * s;
    }
}

// ---------------------------------------------------------------------------
// Kernel 4: scatter-add  out[dst] += h[src]   (8 threads per edge, 4 cols each)
// Coalesced float4 gather of h; 4 non-returning global_atomic_add_f32 per thread.
// out and h both fit in the 192MB L2, so the RMW traffic stays on-chip.
// ---------------------------------------------------------------------------
__global__ void gcn_scatter_kernel(const float* __restrict__ h,
                                   const int* __restrict__ src,
                                   const int* __restrict__ dst,
                                   float* __restrict__ out) {
    long t = (long)blockIdx.x * blockDim.x + threadIdx.x;
    long e = t >> 3;
    if (e >= N_EDGES) return;
    const int cg = (int)(t & 7) * 4;
    const int s = src[e];
    const int d = dst[e];
    const float4 val = *(const float4*)(h + (long)s * HID_DIM + cg);
    float* o = out + (long)d * HID_DIM + cg;
    atomicAdd(o + 0, val.x);
    atomicAdd(o + 1, val.y);
    atomicAdd(o + 2, val.z);
    atomicAdd(o + 3, val.w);
}

// ---------------------------------------------------------------------------
// Kernel 5: out = relu(out * rsqrt(max(deg_in,1)) + b)
// ---------------------------------------------------------------------------
__global__ void gcn_finalize_kernel(float* __restrict__ out,
                                    const float* __restrict__ deg_in,
                                    const float* __restrict__ b) {
    int i = blockIdx.x * blockDim.x + threadIdx.x;
    if (i >= N_NODES * HID_DIM) return;
    const int node = i >> 5;       // HID_DIM == 32
    const int col  = i & 31;
    const float nd = __frsqrt_rn(fmaxf(deg_in[node], 1.0f));
    const float v  = out[i] * nd + b[col];
    out[i] = fmaxf(v, 0.0f);
}

// ---------------------------------------------------------------------------
// Launch: zero -> degrees -> WMMA gemm (+norm_src) -> scatter -> finalize
// d_ws layout: h[N*32] | deg_out[N] | deg_in[N]  = 13.6 MB
// ---------------------------------------------------------------------------
extern "C" void kernel_launch(void* const* d_in, const int* in_sizes, int n_in,
                              void* d_out, int out_size, void* d_ws, size_t ws_size,
                              hipStream_t stream) {
    const float* features = (const float*)d_in[0];
    const int*   src      = (const int*)d_in[1];
    const int*   dst      = (const int*)d_in[2];
    const float* W        = (const float*)d_in[3];
    const float* b        = (const float*)d_in[4];
    float*       out      = (float*)d_out;

    float* h       = (float*)d_ws;                       // N*HID floats
    float* degs    = h + (size_t)N_NODES * HID_DIM;      // 2*N floats
    float* deg_out = degs;
    float* deg_in  = degs + N_NODES;

    const int totZ = N_NODES * (HID_DIM + 2);
    gcn_zero_kernel<<<(totZ + 255) / 256, 256, 0, stream>>>(out, degs);

    gcn_degree_kernel<<<(N_EDGES + 255) / 256, 256, 0, stream>>>(src, dst, deg_out, deg_in);

    gcn_gemm_wmma<<<(N_TILES + 7) / 8, 256, 0, stream>>>(features, W, deg_out, h);

    const long scatThreads = (long)N_EDGES * 8;
    gcn_scatter_kernel<<<(int)((scatThreads + 255) / 256), 256, 0, stream>>>(h, src, dst, out);

    gcn_finalize_kernel<<<(N_NODES * HID_DIM + 255) / 256, 256, 0, stream>>>(out, deg_in, b);
}